// ODEFunction_complex_x_5196910428985
// MI455X (gfx1250) — compile-verified
//
#include <hip/hip_runtime.h>

typedef __attribute__((ext_vector_type(2))) float v2f;
typedef __attribute__((ext_vector_type(8))) float v8f;

#define MODES1            64
#define NROWS             (16384 * 3)          // N * D rows per mode
#define CH                16                   // in/out channels
#define TILES_PER_MODE    (NROWS / 16)         // 3072
#define BLOCKS_PER_MODE   24
#define WAVES_PER_BLOCK   8
#define WAVES_PER_MODE    (BLOCKS_PER_MODE * WAVES_PER_BLOCK)   // 192
#define TILES_PER_WAVE    (TILES_PER_MODE / WAVES_PER_MODE)     // 16 (exact)

__global__ __launch_bounds__(256) void spectral_complex_mul_wmma(
    const float* __restrict__ xr,   // [M, NROWS, 16]
    const float* __restrict__ xi,   // [M, NROWS, 16]
    const float* __restrict__ w,    // [16, 16, M, 2] (re, im)
    float* __restrict__ out)        // [M, NROWS, 16] complex interleaved
{
    const int m     = blockIdx.y;
    const int lane  = threadIdx.x & 31;
    const int wave  = threadIdx.x >> 5;
    const int waveId = blockIdx.x * WAVES_PER_BLOCK + wave;   // 0..191 within mode

    const int n     = lane & 15;    // column for B/C/D, row for A
    const int khalf = lane >> 4;    // 0: K pair {0,1}; 1: K pair {2,3}

    // ---- Load per-mode weights (B operands), reused across all tiles ----
    // complex w[i][o][m] lives at float index ((i*16 + o)*64 + m)*2
    v2f Br[4], Bi[4], Bin[4];
    const float* wb = w + (size_t)m * 2;
#pragma unroll
    for (int kc = 0; kc < 4; ++kc) {
        const int k0 = kc * 4 + khalf * 2;
        v2f c0 = *(const v2f*)(wb + (size_t)((k0      * 16 + n) * 64) * 2);
        v2f c1 = *(const v2f*)(wb + (size_t)(((k0 + 1) * 16 + n) * 64) * 2);
        Br[kc]  = (v2f){c0.x, c1.x};   // element 0 -> B VGPR0 (K=k0), 1 -> VGPR1 (K=k0+1)
        Bi[kc]  = (v2f){c0.y, c1.y};
        Bin[kc] = -Bi[kc];             // pre-negated imag so every WMMA accumulates
    }

    const size_t xbase = (size_t)m * NROWS * CH;

#pragma unroll 1
    for (int tIdx = 0; tIdx < TILES_PER_WAVE; ++tIdx) {
        const int tile = waveId + tIdx * WAVES_PER_MODE;
        const int row0 = tile * 16;

        // ---- Load A tile: lane holds row (lane&15), K pair selected by khalf ----
        const float* ar = xr + xbase + (size_t)(row0 + n) * CH + khalf * 2;
        const float* ai = xi + xbase + (size_t)(row0 + n) * CH + khalf * 2;
        v2f Ar[4], Ai[4];
#pragma unroll
        for (int kc = 0; kc < 4; ++kc) {
            Ar[kc] = *(const v2f*)(ar + kc * 4);
            Ai[kc] = *(const v2f*)(ai + kc * 4);
        }

        // ---- 16x16x16 complex matmul = 16 x V_WMMA_F32_16X16X4_F32 ----
        v8f Cr = {};
        v8f Ci = {};
#pragma unroll
        for (int kc = 0; kc < 4; ++kc) {
            Cr = __builtin_amdgcn_wmma_f32_16x16x4_f32(false, Ar[kc], false, Br[kc],
                                                       (short)0, Cr, false, false);
            Cr = __builtin_amdgcn_wmma_f32_16x16x4_f32(false, Ai[kc], false, Bin[kc],
                                                       (short)0, Cr, false, false);
            Ci = __builtin_amdgcn_wmma_f32_16x16x4_f32(false, Ar[kc], false, Bi[kc],
                                                       (short)0, Ci, false, false);
            Ci = __builtin_amdgcn_wmma_f32_16x16x4_f32(false, Ai[kc], false, Br[kc],
                                                       (short)0, Ci, false, false);
        }

        // ---- Store interleaved complex: VGPR j -> row j (+8 for upper half-wave) ----
        float* ob = out + 2 * (xbase + (size_t)row0 * CH);
#pragma unroll
        for (int j = 0; j < 8; ++j) {
            const int r = j + khalf * 8;
            v2f val = {Cr[j], Ci[j]};
            *(v2f*)(ob + 2 * ((size_t)r * CH + n)) = val;
        }
    }
}

extern "C" void kernel_launch(void* const* d_in, const int* in_sizes, int n_in,
                              void* d_out, int out_size, void* d_ws, size_t ws_size,
                              hipStream_t stream) {
    // inputs (setup_inputs order): t, x_ft_real, x_ft_imag, weights1
    const float* xr = (const float*)d_in[1];
    const float* xi = (const float*)d_in[2];
    const float* w  = (const float*)d_in[3];
    float* out = (float*)d_out;

    dim3 grid(BLOCKS_PER_MODE, MODES1);
    dim3 block(WAVES_PER_BLOCK * 32);
    spectral_complex_mul_wmma<<<grid, block, 0, stream>>>(xr, xi, w, out);
}